// GCNStack_70686571758072
// MI455X (gfx1250) — compile-verified
//
#include <hip/hip_runtime.h>
#include <hip/hip_bf16.h>

typedef __attribute__((ext_vector_type(2))) float v2f;
typedef __attribute__((ext_vector_type(8))) float v8f;

// ---------------- degree kernels ----------------
__global__ void deg_init_kernel(float* __restrict__ deg, int N) {
    int v = blockIdx.x * blockDim.x + threadIdx.x;
    if (v < N) deg[v] = 1.0f;   // self-loop contributes 1
}

__global__ void deg_count_kernel(const long long* __restrict__ ei,
                                 float* __restrict__ deg, int E) {
    int e = blockIdx.x * blockDim.x + threadIdx.x;
    if (e < E) {
        int dst = (int)ei[(size_t)E + e];
        atomicAdd(&deg[dst], 1.0f);
    }
}

__global__ void deg_rsqrt_kernel(float* __restrict__ deg, int N) {
    int v = blockIdx.x * blockDim.x + threadIdx.x;
    if (v < N) deg[v] = rsqrtf(deg[v]);   // deg >= 1 always (self-loop)
}

// ---------------- WMMA f32 GEMM: hp = (X @ W) * dinv[row]; also seeds acc ----
// X: [N,128] row-major, W: [128,64] row-major.
// One wave per 16x16 output tile; K stepped by 4 via V_WMMA_F32_16X16X4_F32.
// A frag (16x4 f32): lanes 0-15 hold row M=lane, K={0,1}; lanes 16-31 K={2,3}.
// B frag (4x16 f32): lanes 0-15 hold col N=lane, K={0,1}; lanes 16-31 K={2,3}.
// C/D (16x16 f32):   VGPR i, lanes 0-15 -> M=i, lanes 16-31 -> M=i+8; N=lane%16.
__global__ void __launch_bounds__(256)
gemm_wmma_kernel(const float* __restrict__ X,
                 const float* __restrict__ W,
                 const float* __restrict__ dinv,
                 float* __restrict__ hp,    // [N,64] workspace (pre-scaled h)
                 float* __restrict__ acc,   // [N,64] accumulator (= emb region)
                 int N) {
    const int wave = threadIdx.x >> 5;
    const int lane = threadIdx.x & 31;
    const int tile = blockIdx.x * 8 + wave;      // 8 waves / block
    const int numRowTiles = (N + 15) >> 4;
    const int rowTile = tile >> 2;               // 4 col tiles (64 cols)
    const int colTile = tile & 3;
    if (rowTile >= numRowTiles) return;          // wave-uniform exit (EXEC all-1)

    const int row0 = rowTile << 4;
    const int col0 = colTile << 4;
    const int half = lane >> 4;                  // 0: K={0,1}, 1: K={2,3}
    const int l16  = lane & 15;

    int arow = row0 + l16; if (arow >= N) arow = N - 1;   // clamp (store guarded)
    const int bcol = col0 + l16;

    const float* xrow = X + (size_t)arow * 128 + (half << 1);
    const float* wcol = W + (size_t)(half << 1) * 64 + bcol;

    v8f c = {};
#pragma unroll
    for (int k0 = 0; k0 < 128; k0 += 4) {
        v2f a = *(const v2f*)(xrow + k0);        // X[arow][k0+2h .. k0+2h+1]
        v2f b;
        b.x = wcol[(size_t)k0 * 64];             // W[k0+2h  ][bcol]
        b.y = wcol[(size_t)k0 * 64 + 64];        // W[k0+2h+1][bcol]
        c = __builtin_amdgcn_wmma_f32_16x16x4_f32(
                /*neg_a=*/false, a, /*neg_b=*/false, b,
                /*c_mod=*/(short)0, c, /*reuse_a=*/false, /*reuse_b=*/false);
    }

#pragma unroll
    for (int i = 0; i < 8; ++i) {
        const int row = row0 + i + (half << 3);
        if (row < N) {
            const float v = c[i] * dinv[row];
            const size_t idx = (size_t)row * 64 + col0 + l16;
            hp[idx]  = v;   // pre-scaled features for gathering
            acc[idx] = v;   // self-loop seed: out[v] gets + hp[v]
        }
    }
}

// ---------------- edge scatter-add: acc[dst] += hp[src] -----------------
// 16 threads per edge, float4 per thread -> E*64 f32 L2 atomics.
__global__ void __launch_bounds__(256)
scatter_kernel(const long long* __restrict__ ei,
               const float* __restrict__ hp,
               float* __restrict__ acc, int E) {
    const long long tid = (long long)blockIdx.x * blockDim.x + threadIdx.x;
    const long long e = tid >> 4;
    if (e >= E) return;
    const int q = (int)(tid & 15);
    const int src = (int)ei[e];
    const int dst = (int)ei[(size_t)E + e];
    const float4 v = *(const float4*)(hp + (size_t)src * 64 + q * 4);
    float* a = acc + (size_t)dst * 64 + q * 4;
    atomicAdd(a + 0, v.x);
    atomicAdd(a + 1, v.y);
    atomicAdd(a + 2, v.z);
    atomicAdd(a + 3, v.w);
}

// ---------------- finalize: emb = dinv*acc + b; logp = log_softmax(relu(emb))
// One wave32 per row; 64 features -> 2 per lane; shfl_xor reductions.
__global__ void __launch_bounds__(256)
finalize_kernel(const float* __restrict__ dinv,
                const float* __restrict__ bias,
                float* __restrict__ emb,    // in: acc, out: emb (in place)
                float* __restrict__ logp,
                int N) {
    const int wave = threadIdx.x >> 5;
    const int lane = threadIdx.x & 31;
    const int row = blockIdx.x * 8 + wave;
    if (row >= N) return;
    const float d = dinv[row];
    const size_t base = (size_t)row * 64;
    const int f0 = lane, f1 = lane + 32;

    const float e0 = emb[base + f0] * d + bias[f0];
    const float e1 = emb[base + f1] * d + bias[f1];
    const float r0 = fmaxf(e0, 0.0f);
    const float r1 = fmaxf(e1, 0.0f);

    float m = fmaxf(r0, r1);
#pragma unroll
    for (int off = 16; off > 0; off >>= 1)
        m = fmaxf(m, __shfl_xor(m, off, 32));
    float s = expf(r0 - m) + expf(r1 - m);
#pragma unroll
    for (int off = 16; off > 0; off >>= 1)
        s += __shfl_xor(s, off, 32);
    const float ls = logf(s) + m;

    emb[base + f0]  = e0;
    emb[base + f1]  = e1;
    logp[base + f0] = r0 - ls;
    logp[base + f1] = r1 - ls;
}

extern "C" void kernel_launch(void* const* d_in, const int* in_sizes, int n_in,
                              void* d_out, int out_size, void* d_ws, size_t ws_size,
                              hipStream_t stream) {
    const float*      x  = (const float*)d_in[0];       // [N,128]
    const long long*  ei = (const long long*)d_in[1];   // [2,E] int64
    const float*      W  = (const float*)d_in[3];       // [128,64]
    const float*      b  = (const float*)d_in[4];       // [64]

    const int N = in_sizes[0] / 128;
    const int E = in_sizes[1] / 2;

    float* out  = (float*)d_out;
    float* emb  = out;                      // [N,64] also used as accumulator
    float* logp = out + (size_t)N * 64;     // [N,64]

    float* deg = (float*)d_ws;              // [N] degree -> dinv (rsqrt in place)
    float* hp  = deg + N;                   // [N,64] pre-scaled features

    const int T = 256;
    deg_init_kernel<<<(N + T - 1) / T, T, 0, stream>>>(deg, N);
    deg_count_kernel<<<(E + T - 1) / T, T, 0, stream>>>(ei, deg, E);
    deg_rsqrt_kernel<<<(N + T - 1) / T, T, 0, stream>>>(deg, N);

    const int tiles = ((N + 15) / 16) * 4;          // 16x16 tiles over [N,64]
    gemm_wmma_kernel<<<(tiles + 7) / 8, T, 0, stream>>>(x, W, deg, hp, emb, N);

    const long long sthreads = (long long)E * 16;
    scatter_kernel<<<(unsigned)((sthreads + T - 1) / T), T, 0, stream>>>(ei, hp, emb, E);

    finalize_kernel<<<(N + 7) / 8, T, 0, stream>>>(deg, b, emb, logp, N);
}